// UpdateNodeBlock_6734508720700
// MI455X (gfx1250) — compile-verified
//
#include <hip/hip_runtime.h>
#include <math.h>

#define N_ATOMS_ 50000
#define N_EDGES_ 800000
#define C_ 64
#define N_BASIS_ 16
#define INV_NORM_ 0.0625f   // 1/16
#define EPS_ 1e-8f

typedef __attribute__((ext_vector_type(2))) float v2f;
typedef __attribute__((ext_vector_type(8))) float v8f;

// ---------------------------------------------------------------------------
// Kernel 1: zero the scatter accumulators in workspace (harness poisons ws).
// ---------------------------------------------------------------------------
__global__ __launch_bounds__(256) void zero_kernel(float* __restrict__ p, int n) {
    int i = blockIdx.x * 256 + threadIdx.x;
    if (i < n) p[i] = 0.0f;
}

// ---------------------------------------------------------------------------
// Kernel 2: fused GraphConv edge kernel.
// One thread per (edge, channel). Radial-filter weights staged in LDS (16KB).
// Scatter-add via global_atomic_add_f32 (resolves in L2; node accumulators
// are 51 MB total, L2 is 192 MB).
// ---------------------------------------------------------------------------
__global__ __launch_bounds__(256) void edge_kernel(
    const float* __restrict__ node_0, const float* __restrict__ node_1,
    const float* __restrict__ rbf,    const float* __restrict__ rhat,
    const int*   __restrict__ idx_i,  const int*   __restrict__ idx_j,
    const float* __restrict__ W00, const float* __restrict__ W01,
    const float* __restrict__ W10, const float* __restrict__ W11,
    float* __restrict__ r0, float* __restrict__ r1)
{
    __shared__ float sW[4 * N_BASIS_ * C_];   // 4 x 16 x 64 floats = 16 KB
    for (int t = threadIdx.x; t < N_BASIS_ * C_; t += 256) {
        sW[0 * 1024 + t] = W00[t];
        sW[1 * 1024 + t] = W01[t];
        sW[2 * 1024 + t] = W10[t];
        sW[3 * 1024 + t] = W11[t];
    }
    __syncthreads();

    int gid = blockIdx.x * 256 + threadIdx.x;          // < 51.2M
    if (gid >= N_EDGES_ * C_) return;
    int e = gid >> 6;
    int c = gid & 63;

    int i = idx_i[e];
    int j = idx_j[e];

    // radial filters for this (edge, channel): 16-term dots against LDS weights
    float f00 = 0.f, f01 = 0.f, f10 = 0.f, f11 = 0.f;
#pragma unroll
    for (int b = 0; b < N_BASIS_; ++b) {
        float rb = rbf[e * N_BASIS_ + b];
        int   wi = b * C_ + c;
        f00 += rb * sW[0 * 1024 + wi];
        f01 += rb * sW[1 * 1024 + wi];
        f10 += rb * sW[2 * 1024 + wi];
        f11 += rb * sW[3 * 1024 + wi];
    }

    float rx = rhat[e * 3 + 0];
    float ry = rhat[e * 3 + 1];
    float rz = rhat[e * 3 + 2];

    float n0  = node_0[j * C_ + c];
    int   b1  = (j * C_ + c) * 3;
    float n1x = node_1[b1 + 0];
    float n1y = node_1[b1 + 1];
    float n1z = node_1[b1 + 2];

    float dot = n1x * rx + n1y * ry + n1z * rz;
    float m0  = n0 * f00 + f11 * dot;
    float g   = n0 * f01;

    atomicAdd(&r0[i * C_ + c], m0);
    int ob = (i * C_ + c) * 3;
    atomicAdd(&r1[ob + 0], n1x * f10 + g * rx);
    atomicAdd(&r1[ob + 1], n1y * f10 + g * ry);
    atomicAdd(&r1[ob + 2], n1z * f10 + g * rz);
}

// ---------------------------------------------------------------------------
// Kernel 3: SelfInteraction (WMMA f32 16x16x4) + silu / norm-gate + residual.
// Block = 128 threads = 4 waves; block handles 16 nodes; wave w handles
// output-channel tile w (16 channels). Each wave carries 4 accumulators
// (s0, s1x, s1y, s1z) sharing one C/D layout, so the equivariant gate is
// pure elementwise epilogue.
// ---------------------------------------------------------------------------
__device__ inline v8f wmma_k4(v2f a, v2f b, v8f c) {
    return __builtin_amdgcn_wmma_f32_16x16x4_f32(
        false, a, false, b, (short)0, c, false, false);
}

__global__ __launch_bounds__(128) void si_wmma_kernel(
    const float* __restrict__ r0,   const float* __restrict__ r1,
    const float* __restrict__ Wsi0, const float* __restrict__ bsi0,
    const float* __restrict__ Wsi1,
    const float* __restrict__ node_0, const float* __restrict__ node_1,
    float* __restrict__ out0, float* __restrict__ out1)
{
    int tile      = blockIdx.x;            // node tile (16 nodes); 3125 exact
    int wave      = threadIdx.x >> 5;      // 0..3 -> output channel tile
    int lane      = threadIdx.x & 31;
    int node_base = tile * 16;
    int m_ab      = lane & 15;             // A row / B col within tile
    int khalf     = (lane >> 4) ? 2 : 0;   // upper half-wave holds k+2,k+3
    int ncol      = wave * 16 + (lane & 15);

    v8f acc0 = {}; v8f accx = {}; v8f accy = {}; v8f accz = {};

    for (int ks = 0; ks < 16; ++ks) {
        int k0 = ks * 4 + khalf;

        // A fragments: rows = nodes, k = input channel; fold the 1/16 norm in.
        int a0i = (node_base + m_ab) * C_ + k0;
        v2f a0; a0.x = r0[a0i]     * INV_NORM_;
                a0.y = r0[a0i + 1] * INV_NORM_;
        int a1i = a0i * 3;                       // r1[node, k, comp]
        v2f ax; ax.x = r1[a1i + 0] * INV_NORM_; ax.y = r1[a1i + 3] * INV_NORM_;
        v2f ay; ay.x = r1[a1i + 1] * INV_NORM_; ay.y = r1[a1i + 4] * INV_NORM_;
        v2f az; az.x = r1[a1i + 2] * INV_NORM_; az.y = r1[a1i + 5] * INV_NORM_;

        // B fragments: W[k][outchan], same half-wave k split as A.
        v2f b0; b0.x = Wsi0[k0 * C_ + ncol]; b0.y = Wsi0[(k0 + 1) * C_ + ncol];
        v2f b1; b1.x = Wsi1[k0 * C_ + ncol]; b1.y = Wsi1[(k0 + 1) * C_ + ncol];

        acc0 = wmma_k4(a0, b0, acc0);
        accx = wmma_k4(ax, b1, accx);
        accy = wmma_k4(ay, b1, accy);
        accz = wmma_k4(az, b1, accz);
    }

    float bias = bsi0[ncol];

#pragma unroll
    for (int r = 0; r < 8; ++r) {
        int m    = r + ((lane >> 4) ? 8 : 0);     // C/D row for this vgpr
        int node = node_base + m;

        float s0 = acc0[r] + bias;
        float o0 = s0 / (1.0f + __expf(-s0));     // silu
        int   oi = node * C_ + ncol;
        out0[oi] = node_0[oi] + o0;

        float sx = accx[r], sy = accy[r], sz = accz[r];
        float nrm  = sqrtf(sx * sx + sy * sy + sz * sz + EPS_);
        float gate = nrm / (1.0f + __expf(-nrm)); // silu(norm)
        int   ob   = oi * 3;
        out1[ob + 0] = node_1[ob + 0] + sx * gate;
        out1[ob + 1] = node_1[ob + 1] + sy * gate;
        out1[ob + 2] = node_1[ob + 2] + sz * gate;
    }
}

// ---------------------------------------------------------------------------
extern "C" void kernel_launch(void* const* d_in, const int* in_sizes, int n_in,
                              void* d_out, int out_size, void* d_ws, size_t ws_size,
                              hipStream_t stream) {
    const float* node_0 = (const float*)d_in[0];
    const float* node_1 = (const float*)d_in[1];
    const float* rbf    = (const float*)d_in[2];
    const float* rhat   = (const float*)d_in[3];
    const int*   idx_i  = (const int*)d_in[4];
    const int*   idx_j  = (const int*)d_in[5];
    const float* W00    = (const float*)d_in[6];
    const float* W01    = (const float*)d_in[7];
    const float* W10    = (const float*)d_in[8];
    const float* W11    = (const float*)d_in[9];
    const float* Wsi0   = (const float*)d_in[10];
    const float* bsi0   = (const float*)d_in[11];
    const float* Wsi1   = (const float*)d_in[12];

    float* r0   = (float*)d_ws;                       // [N, C]
    float* r1   = r0 + (size_t)N_ATOMS_ * C_;         // [N, C, 3]
    float* out0 = (float*)d_out;                      // [N, C]
    float* out1 = out0 + (size_t)N_ATOMS_ * C_;       // [N, C, 3]

    int accN = N_ATOMS_ * C_ * 4;                     // 12.8M floats
    hipLaunchKernelGGL(zero_kernel, dim3((accN + 255) / 256), dim3(256), 0, stream,
                       r0, accN);

    int totalEC = N_EDGES_ * C_;                      // 51.2M threads
    hipLaunchKernelGGL(edge_kernel, dim3((totalEC + 255) / 256), dim3(256), 0, stream,
                       node_0, node_1, rbf, rhat, idx_i, idx_j,
                       W00, W01, W10, W11, r0, r1);

    hipLaunchKernelGGL(si_wmma_kernel, dim3(N_ATOMS_ / 16), dim3(128), 0, stream,
                       r0, r1, Wsi0, bsi0, Wsi1, node_0, node_1, out0, out1);
}